// DrugGIN_65841848648261
// MI455X (gfx1250) — compile-verified
//
#include <hip/hip_runtime.h>
#include <hip/hip_bf16.h>

#define N_NODES  100000
#define N_EDGES  1600000
#define N_GRAPHS 2048
#define BN_EPS   1e-5f

typedef __attribute__((ext_vector_type(2))) float v2f;
typedef __attribute__((ext_vector_type(8))) float v8f;

// ---------------- elementwise helpers ----------------

__global__ void zero_f32(float* __restrict__ p, long n) {
    long i = (long)blockIdx.x * blockDim.x + threadIdx.x;
    if (i < n) p[i] = 0.f;
}

__global__ void copy_f32(const float* __restrict__ s, float* __restrict__ d, long n) {
    long i = (long)blockIdx.x * blockDim.x + threadIdx.x;
    if (i < n) d[i] = s[i];
}

// agg[dst[e]][f] += X[src[e]][f]  (scalar feature path, D arbitrary)
__global__ void edge_scatter(const float* __restrict__ X,
                             const long long* __restrict__ src,
                             const long long* __restrict__ dst,
                             float* __restrict__ agg, int D) {
    long i = (long)blockIdx.x * blockDim.x + threadIdx.x;
    long total = (long)N_EDGES * D;
    if (i >= total) return;
    long e = i / D;
    int  f = (int)(i - e * (long)D);
    long s = src[e], t = dst[e];
    atomicAdd(&agg[t * (long)D + f], X[s * (long)D + f]);
}

// vectorized path for D % 4 == 0: one float4 load + 4 atomics per thread
__global__ void edge_scatter4(const float* __restrict__ X,
                              const long long* __restrict__ src,
                              const long long* __restrict__ dst,
                              float* __restrict__ agg, int D4) {  // D4 = D/4
    long i = (long)blockIdx.x * blockDim.x + threadIdx.x;
    long total = (long)N_EDGES * D4;
    if (i >= total) return;
    long e = i / D4;
    int  f4 = (int)(i - e * (long)D4);
    long s = src[e], t = dst[e];
    const float4 v = ((const float4*)X)[s * (long)D4 + f4];
    float* ap = &agg[(t * (long)D4 + f4) * 4];
    atomicAdd(ap + 0, v.x);
    atomicAdd(ap + 1, v.y);
    atomicAdd(ap + 2, v.z);
    atomicAdd(ap + 3, v.w);
}

// ---------------- fp32 WMMA GEMM: C[M,N] = A[M,K] @ W[K,N] + bias, optional ReLU ----
// One wave per 16x64 output tile: 4 accumulators share a single A fragment per
// k-step (4x V_WMMA_F32_16X16X4_F32 per iteration). Requires M%16==0, N%64==0.

__global__ void gemm_wmma_f32(const float* __restrict__ A,
                              const float* __restrict__ W,
                              const float* __restrict__ bias,
                              float* __restrict__ C,
                              int M, int K, int N, int relu) {
    const int wave = threadIdx.x >> 5;
    const int lane = threadIdx.x & 31;
    const int half = lane >> 4;
    const int idx  = lane & 15;
    const int tiles_n = N >> 6;                               // 64-wide N tiles
    const int tiles_m = M >> 4;
    const int tile = blockIdx.x * (blockDim.x >> 5) + wave;   // wave-uniform
    if (tile >= tiles_m * tiles_n) return;                    // wave-uniform exit
    const int mt = tile / tiles_n;
    const int nt = tile - mt * tiles_n;
    const int m0 = mt << 4, n0 = nt << 6;

    const float* Arow = A + (long)(m0 + idx) * K;   // this lane's A row
    const float* Wcol = W + n0 + idx;               // this lane's B column base
    const int kh = 2 * half;

    v8f acc0 = {0.f,0.f,0.f,0.f,0.f,0.f,0.f,0.f};
    v8f acc1 = acc0, acc2 = acc0, acc3 = acc0;

    int k0 = 0;
    for (; k0 + 4 <= K; k0 += 4) {
        v2f a;
        a[0] = Arow[k0 + kh];
        a[1] = Arow[k0 + kh + 1];
        const float* w0 = Wcol + (long)(k0 + kh) * N;
        const float* w1 = Wcol + (long)(k0 + kh + 1) * N;
        v2f b0, b1, b2, b3;
        b0[0] = w0[0];   b0[1] = w1[0];
        b1[0] = w0[16];  b1[1] = w1[16];
        b2[0] = w0[32];  b2[1] = w1[32];
        b3[0] = w0[48];  b3[1] = w1[48];
        acc0 = __builtin_amdgcn_wmma_f32_16x16x4_f32(false, a, false, b0, (short)0, acc0, false, false);
        acc1 = __builtin_amdgcn_wmma_f32_16x16x4_f32(false, a, false, b1, (short)0, acc1, false, false);
        acc2 = __builtin_amdgcn_wmma_f32_16x16x4_f32(false, a, false, b2, (short)0, acc2, false, false);
        acc3 = __builtin_amdgcn_wmma_f32_16x16x4_f32(false, a, false, b3, (short)0, acc3, false, false);
    }
    if (k0 < K) {   // K % 4 != 0 tail (layer 0: K = 66), guard each element
        const int ka = k0 + kh, kb = k0 + kh + 1;
        const bool va = (ka < K), vb = (kb < K);
        v2f a;
        a[0] = va ? Arow[ka] : 0.f;
        a[1] = vb ? Arow[kb] : 0.f;
        const float* w0 = Wcol + (long)ka * N;
        const float* w1 = Wcol + (long)kb * N;
        v2f b0, b1, b2, b3;
        b0[0] = va ? w0[0]  : 0.f;  b0[1] = vb ? w1[0]  : 0.f;
        b1[0] = va ? w0[16] : 0.f;  b1[1] = vb ? w1[16] : 0.f;
        b2[0] = va ? w0[32] : 0.f;  b2[1] = vb ? w1[32] : 0.f;
        b3[0] = va ? w0[48] : 0.f;  b3[1] = vb ? w1[48] : 0.f;
        acc0 = __builtin_amdgcn_wmma_f32_16x16x4_f32(false, a, false, b0, (short)0, acc0, false, false);
        acc1 = __builtin_amdgcn_wmma_f32_16x16x4_f32(false, a, false, b1, (short)0, acc1, false, false);
        acc2 = __builtin_amdgcn_wmma_f32_16x16x4_f32(false, a, false, b2, (short)0, acc2, false, false);
        acc3 = __builtin_amdgcn_wmma_f32_16x16x4_f32(false, a, false, b3, (short)0, acc3, false, false);
    }

    const float bv0 = bias[n0 + idx];
    const float bv1 = bias[n0 + 16 + idx];
    const float bv2 = bias[n0 + 32 + idx];
    const float bv3 = bias[n0 + 48 + idx];
#pragma unroll
    for (int v = 0; v < 8; ++v) {
        const int row = m0 + v + 8 * half;
        float* crow = C + (long)row * N + n0 + idx;
        float v0 = acc0[v] + bv0;
        float v1 = acc1[v] + bv1;
        float v2 = acc2[v] + bv2;
        float v3 = acc3[v] + bv3;
        if (relu) {
            v0 = fmaxf(v0, 0.f); v1 = fmaxf(v1, 0.f);
            v2 = fmaxf(v2, 0.f); v3 = fmaxf(v3, 0.f);
        }
        crow[0]  = v0;
        crow[16] = v1;
        crow[32] = v2;
        crow[48] = v3;
    }
}

// ---------------- BatchNorm pieces ----------------

// blockDim.x == N (128 or 256); each thread owns a column, accumulates a row slab
// in registers, one pair of atomics per block.
__global__ void col_stats(const float* __restrict__ H,
                          float* __restrict__ sum, float* __restrict__ sumsq,
                          int M, int N, int rows_per_block) {
    const int c  = threadIdx.x;
    const int r0 = blockIdx.x * rows_per_block;
    int r1 = r0 + rows_per_block; if (r1 > M) r1 = M;
    float s = 0.f, s2 = 0.f;
    for (int r = r0; r < r1; ++r) {
        const float v = H[(long)r * N + c];
        s += v; s2 += v * v;
    }
    atomicAdd(&sum[c], s);
    atomicAdd(&sumsq[c], s2);
}

__global__ void bn_params(const float* __restrict__ sum, const float* __restrict__ sumsq,
                          const float* __restrict__ gamma, const float* __restrict__ beta,
                          float* __restrict__ scale, float* __restrict__ shift,
                          int N, float invM) {
    const int c = blockIdx.x * blockDim.x + threadIdx.x;
    if (c >= N) return;
    const float mean = sum[c] * invM;
    const float var  = sumsq[c] * invM - mean * mean;
    const float sc   = gamma[c] * rsqrtf(var + BN_EPS);
    scale[c] = sc;
    shift[c] = beta[c] - mean * sc;
}

__global__ void bn_relu(float* __restrict__ H, const float* __restrict__ scale,
                        const float* __restrict__ shift, long total, int N) {
    long i = (long)blockIdx.x * blockDim.x + threadIdx.x;
    if (i >= total) return;
    const int c = (int)(i & (long)(N - 1));   // N is a power of two (128/256)
    H[i] = fmaxf(H[i] * scale[c] + shift[c], 0.f);
}

// ---------------- global add pool ----------------

__global__ void pool_add(const float* __restrict__ H,
                         const long long* __restrict__ batch,
                         float* __restrict__ G, int N) {
    long i = (long)blockIdx.x * blockDim.x + threadIdx.x;
    long total = (long)N_NODES * N;
    if (i >= total) return;
    long node = i / N;
    int  f    = (int)(i - node * (long)N);
    atomicAdd(&G[batch[node] * (long)N + f], H[i]);
}

// ---------------- driver ----------------

static inline long cdiv(long a, long b) { return (a + b - 1) / b; }

extern "C" void kernel_launch(void* const* d_in, const int* in_sizes, int n_in,
                              void* d_out, int out_size, void* d_ws, size_t ws_size,
                              hipStream_t stream) {
    (void)in_sizes; (void)n_in; (void)out_size; (void)ws_size;

    const float*     x     = (const float*)d_in[0];
    const long long* ei    = (const long long*)d_in[1];
    const long long* src   = ei;
    const long long* dst   = ei + N_EDGES;
    const long long* batch = (const long long*)d_in[2];

    const float* P[25];
    for (int i = 3; i < 25; ++i) P[i] = (const float*)d_in[i];
    const float* fc0_w = P[21]; const float* fc0_b = P[22];
    const float* fc1_w = P[23]; const float* fc1_b = P[24];

    // workspace carve-up (floats)
    float* buf0 = (float*)d_ws;                         // node features  100000*256
    float* buf1 = buf0 + (size_t)N_NODES * 256;         // agg            100000*256
    float* buf2 = buf1 + (size_t)N_NODES * 256;         // gemm1 out      100000*256
    float* g    = buf2 + (size_t)N_NODES * 256;         // pooled         2048*128
    float* hid  = g    + (size_t)N_GRAPHS * 128;        // fc hidden      2048*1024
    float* sum  = hid  + (size_t)N_GRAPHS * 1024;       // 512
    float* sumsq = sum + 512;                           // 512
    float* scale = sumsq + 512;                         // 256
    float* shift = scale + 256;                         // 256

    auto run_layer = [&](const float* Hin, int Din, int Dout,
                         const float* w1, const float* b1,
                         const float* gamma, const float* beta,
                         const float* w2, const float* b2,
                         float* Hout) {
        // agg = Hin + scatter-add over edges
        const long nIn = (long)N_NODES * Din;
        copy_f32<<<cdiv(nIn, 256), 256, 0, stream>>>(Hin, buf1, nIn);
        if ((Din & 3) == 0) {
            const long nE4 = (long)N_EDGES * (Din >> 2);
            edge_scatter4<<<cdiv(nE4, 256), 256, 0, stream>>>(Hin, src, dst, buf1, Din >> 2);
        } else {
            const long nE = (long)N_EDGES * Din;
            edge_scatter<<<cdiv(nE, 256), 256, 0, stream>>>(Hin, src, dst, buf1, Din);
        }

        // buf2 = agg @ w1 + b1
        const long tiles1 = (long)(N_NODES / 16) * (Dout / 64);
        gemm_wmma_f32<<<cdiv(tiles1, 8), 256, 0, stream>>>(buf1, w1, b1, buf2,
                                                           N_NODES, Din, Dout, 0);
        // batch-norm statistics over node axis
        zero_f32<<<4, 256, 0, stream>>>(sum, 1024);     // clears sum+sumsq
        const int rpb = 256;
        col_stats<<<cdiv(N_NODES, rpb), Dout, 0, stream>>>(buf2, sum, sumsq,
                                                           N_NODES, Dout, rpb);
        bn_params<<<1, Dout, 0, stream>>>(sum, sumsq, gamma, beta, scale, shift,
                                          Dout, 1.0f / (float)N_NODES);
        const long tot = (long)N_NODES * Dout;
        bn_relu<<<cdiv(tot, 256), 256, 0, stream>>>(buf2, scale, shift, tot, Dout);

        // Hout = relu(buf2 @ w2 + b2)
        gemm_wmma_f32<<<cdiv(tiles1, 8), 256, 0, stream>>>(buf2, w2, b2, Hout,
                                                           N_NODES, Dout, Dout, 1);
    };

    // dims: (66,128), (128,256), (256,128)
    run_layer(x,    66,  128, P[3],  P[4],  P[5],  P[6],  P[7],  P[8],  buf0);
    run_layer(buf0, 128, 256, P[9],  P[10], P[11], P[12], P[13], P[14], buf0);
    run_layer(buf0, 256, 128, P[15], P[16], P[17], P[18], P[19], P[20], buf0);

    // global add pool -> g[2048,128]
    zero_f32<<<cdiv((long)N_GRAPHS * 128, 256), 256, 0, stream>>>(g, (long)N_GRAPHS * 128);
    pool_add<<<cdiv((long)N_NODES * 128, 256), 256, 0, stream>>>(buf0, batch, g, 128);

    // hid = relu(g @ fc0_w + fc0_b)   [2048,128]x[128,1024]
    const long tf0 = (long)(N_GRAPHS / 16) * (1024 / 64);
    gemm_wmma_f32<<<cdiv(tf0, 8), 256, 0, stream>>>(g, fc0_w, fc0_b, hid,
                                                    N_GRAPHS, 128, 1024, 1);
    // out = hid @ fc1_w + fc1_b       [2048,1024]x[1024,128]
    const long tf1 = (long)(N_GRAPHS / 16) * (128 / 64);
    gemm_wmma_f32<<<cdiv(tf1, 8), 256, 0, stream>>>(hid, fc1_w, fc1_b, (float*)d_out,
                                                    N_GRAPHS, 1024, 128, 0);
}